// SimpleNN_58067957842264
// MI455X (gfx1250) — compile-verified
//
#include <hip/hip_runtime.h>
#include <hip/hip_bf16.h>

// ---------------- problem constants (from reference) ----------------
#define BB   128
#define VV   8192
#define EE   128
#define DEMM 2
#define SRC_LD (DEMM + VV)     // 8194
#define IN_RAW 258             // DEM + 2*E
#define KP1    260             // padded to multiple of 4 for f32 WMMA (K-step 4)
#define H1     512
#define H2     1024
#define H3     1024
#define NOUT   20
#define NOUTP  32              // padded to 2 N-tiles of 16
#define EPSBN  1e-5f

typedef __attribute__((ext_vector_type(2))) float v2f;
typedef __attribute__((ext_vector_type(8))) float v8f;

// ------------------------------------------------------------------
// Kernel 1: sparse embedding aggregation (mean + masked max) -> padded X
// One block per batch row, 128 threads. Deterministic: active indices
// compacted in ascending v order via per-thread counts + prefix scan.
// ------------------------------------------------------------------
__global__ __launch_bounds__(128) void agg_embed_kernel(
    const float* __restrict__ src, const float* __restrict__ embed,
    float* __restrict__ xp) {
  __shared__ int s_idx[VV];
  __shared__ int s_cnt[128];
  __shared__ int s_off[129];

  const int b = blockIdx.x;
  const int t = threadIdx.x;               // 0..127
  const float* codes = src + (size_t)b * SRC_LD + DEMM;

  // phase 1a: count actives in my contiguous chunk of 64 codes
  const int v0 = t * (VV / 128);
  int c = 0;
  for (int v = v0; v < v0 + (VV / 128); ++v)
    c += (codes[v] > 0.0f) ? 1 : 0;
  s_cnt[t] = c;
  __syncthreads();

  // phase 1b: exclusive prefix scan (tiny, serial on thread 0)
  if (t == 0) {
    int acc = 0;
    for (int i = 0; i < 128; ++i) { s_off[i] = acc; acc += s_cnt[i]; }
    s_off[128] = acc;
  }
  __syncthreads();

  // phase 1c: write my actives at my offset (globally sorted by v)
  int p = s_off[t];
  for (int v = v0; v < v0 + (VV / 128); ++v)
    if (codes[v] > 0.0f) s_idx[p++] = v;
  __syncthreads();

  // phase 2: each thread owns embedding column e = t
  const int n = s_off[128];                // >= 1 (codes[:,0] forced to 1)
  float sum = 0.0f;
  float mx  = -3.4028234663852886e38f;     // finfo(float32).min
  for (int i = 0; i < n; ++i) {
    const float val = embed[(size_t)s_idx[i] * EE + t];
    sum += val;
    mx = fmaxf(mx, val);
  }

  float* xr = xp + (size_t)b * KP1;
  if (t < DEMM)  xr[t] = src[(size_t)b * SRC_LD + t];
  xr[DEMM + t]       = sum * (1.0f / (float)n);   // mean_emb
  xr[DEMM + EE + t]  = mx;                        // max_emb
  if (t < KP1 - IN_RAW) xr[IN_RAW + t] = 0.0f;    // K padding
}

// ------------------------------------------------------------------
// Prep kernels: weight padding + fused BN coefficients
// ------------------------------------------------------------------
__global__ void pad_w1_kernel(const float* __restrict__ W1, float* __restrict__ W1p) {
  int i = blockIdx.x * blockDim.x + threadIdx.x;       // over KP1*H1
  if (i < KP1 * H1) {
    int r = i / H1;
    W1p[i] = (r < IN_RAW) ? W1[i] : 0.0f;
  }
}

__global__ void pad_w4_kernel(const float* __restrict__ W4, float* __restrict__ W4p) {
  int i = blockIdx.x * blockDim.x + threadIdx.x;       // over H3*NOUTP
  if (i < H3 * NOUTP) {
    int r = i >> 5, cc = i & (NOUTP - 1);
    W4p[i] = (cc < NOUT) ? W4[r * NOUT + cc] : 0.0f;
  }
}

// bn(g,be,m,v) applied to (dot + b):  alpha*dot + (alpha*(b-m)+be)
__global__ void bn_coeffs_kernel(const float* __restrict__ bias,
                                 const float* __restrict__ g,
                                 const float* __restrict__ be,
                                 const float* __restrict__ m,
                                 const float* __restrict__ v,
                                 float* __restrict__ s, float* __restrict__ t, int n) {
  int i = blockIdx.x * blockDim.x + threadIdx.x;
  if (i < n) {
    float a = g[i] * rsqrtf(v[i] + EPSBN);
    s[i] = a;
    t[i] = a * (bias[i] - m[i]) + be[i];
  }
}

__global__ void pad_bias4_kernel(const float* __restrict__ b4,
                                 float* __restrict__ s4, float* __restrict__ t4) {
  int i = threadIdx.x;                                  // 32 threads
  s4[i] = 1.0f;
  t4[i] = (i < NOUT) ? b4[i] : 0.0f;
}

// ------------------------------------------------------------------
// fp32 WMMA GEMM: C(M x ncols) = epilogue( A(MxK) @ W(KxN) )
// One wave -> one 16x16 tile, K-stepped by 4 with V_WMMA_F32_16X16X4_F32.
// A layout (ISA 7.12.2, 32-bit A 16x4): lanes 0-15 M=rows, VGPR0/1=K0/K1;
//   lanes 16-31 carry K2/K3. B mirrors with N across lanes; C/D: lanes
//   0-15 N, VGPR r -> M=r (halves 0-7 / 8-15).
// ------------------------------------------------------------------
template <bool RELU>
__global__ __launch_bounds__(256) void gemm_wmma_f32_kernel(
    const float* __restrict__ A, const float* __restrict__ W,
    const float* __restrict__ scale, const float* __restrict__ bias,
    float* __restrict__ C, int M, int N, int K, int ldc, int ncols) {
  const int lane = threadIdx.x & 31;
  const int wave = threadIdx.x >> 5;
  const int tiles_n = N >> 4;
  const int tile = blockIdx.x * 8 + wave;               // wave-uniform
  if (tile >= (M >> 4) * tiles_n) return;               // whole wave exits
  const int tm = tile / tiles_n;
  const int tn = tile - tm * tiles_n;

  const int l = lane & 15;                              // row/col within tile
  const int h = lane >> 4;                              // K half select
  const float* __restrict__ Ap = A + (size_t)(tm * 16 + l) * K + h * 2;
  const float* __restrict__ Wp = W + (size_t)(h * 2) * N + tn * 16 + l;

  __builtin_prefetch(Ap, 0, 1);
  __builtin_prefetch(Wp, 0, 1);

  v8f c = {};
  for (int k = 0; k < K; k += 4) {
    v2f a; a.x = Ap[k];               a.y = Ap[k + 1];          // contiguous pair
    v2f b; b.x = Wp[(size_t)k * N];   b.y = Wp[(size_t)k * N + N];
    c = __builtin_amdgcn_wmma_f32_16x16x4_f32(
        /*neg_a=*/false, a, /*neg_b=*/false, b,
        /*c_mod=*/(short)0, c, /*reuse_a=*/false, /*reuse_b=*/false);
  }

  const int col = tn * 16 + l;
  const float s = scale[col];
  const float t = bias[col];
  if (col < ncols) {
    const int row0 = tm * 16 + h * 8;
#pragma unroll
    for (int r = 0; r < 8; ++r) {
      float y = c[r] * s + t;
      if (RELU) y = fmaxf(y, 0.0f);
      C[(size_t)(row0 + r) * ldc + col] = y;
    }
  }
}

// ------------------------------------------------------------------
extern "C" void kernel_launch(void* const* d_in, const int* in_sizes, int n_in,
                              void* d_out, int out_size, void* d_ws, size_t ws_size,
                              hipStream_t stream) {
  const float* src = (const float*)d_in[0];
  const float* emb = (const float*)d_in[1];
  const float* W1  = (const float*)d_in[2];
  const float* b1  = (const float*)d_in[3];
  const float* g1  = (const float*)d_in[4];
  const float* be1 = (const float*)d_in[5];
  const float* m1  = (const float*)d_in[6];
  const float* v1  = (const float*)d_in[7];
  const float* W2  = (const float*)d_in[8];
  const float* b2  = (const float*)d_in[9];
  const float* g2  = (const float*)d_in[10];
  const float* be2 = (const float*)d_in[11];
  const float* m2  = (const float*)d_in[12];
  const float* v2  = (const float*)d_in[13];
  const float* W3  = (const float*)d_in[14];
  const float* b3  = (const float*)d_in[15];
  const float* g3  = (const float*)d_in[16];
  const float* be3 = (const float*)d_in[17];
  const float* m3  = (const float*)d_in[18];
  const float* v3  = (const float*)d_in[19];
  const float* W4  = (const float*)d_in[20];
  const float* b4  = (const float*)d_in[21];
  float* out = (float*)d_out;

  // workspace carve-up (floats)
  float* ws  = (float*)d_ws;
  float* Xp  = ws;                 ws += BB * KP1;       // 128 x 260
  float* W1p = ws;                 ws += KP1 * H1;       // 260 x 512
  float* W4p = ws;                 ws += H3 * NOUTP;     // 1024 x 32
  float* s1  = ws;                 ws += H1;
  float* t1  = ws;                 ws += H1;
  float* s2  = ws;                 ws += H2;
  float* t2  = ws;                 ws += H2;
  float* s3  = ws;                 ws += H3;
  float* t3  = ws;                 ws += H3;
  float* s4  = ws;                 ws += NOUTP;
  float* t4  = ws;                 ws += NOUTP;
  float* h1  = ws;                 ws += BB * H1;        // 128 x 512
  float* h2  = ws;                 ws += BB * H2;        // 128 x 1024
  float* h3  = ws;                 ws += BB * H3;        // 128 x 1024

  // stage 0: prep (independent of aggregation; all on one stream)
  pad_w1_kernel<<<(KP1 * H1 + 255) / 256, 256, 0, stream>>>(W1, W1p);
  pad_w4_kernel<<<(H3 * NOUTP + 255) / 256, 256, 0, stream>>>(W4, W4p);
  bn_coeffs_kernel<<<(H1 + 255) / 256, 256, 0, stream>>>(b1, g1, be1, m1, v1, s1, t1, H1);
  bn_coeffs_kernel<<<(H2 + 255) / 256, 256, 0, stream>>>(b2, g2, be2, m2, v2, s2, t2, H2);
  bn_coeffs_kernel<<<(H3 + 255) / 256, 256, 0, stream>>>(b3, g3, be3, m3, v3, s3, t3, H3);
  pad_bias4_kernel<<<1, NOUTP, 0, stream>>>(b4, s4, t4);

  // stage 1: sparse mean/max embedding aggregation -> Xp (128 x 260)
  agg_embed_kernel<<<BB, 128, 0, stream>>>(src, emb, Xp);

  // stage 2-5: fused GEMM + BN + ReLU chain on the WMMA pipe (fp32)
  // tiles: (M/16)*(N/16); 8 waves (one tile each) per 256-thread block
  gemm_wmma_f32_kernel<true ><<<(8 * (H1    / 16)) / 8, 256, 0, stream>>>(
      Xp, W1p, s1, t1, h1, BB, H1, KP1, H1, H1);
  gemm_wmma_f32_kernel<true ><<<(8 * (H2    / 16)) / 8, 256, 0, stream>>>(
      h1, W2,  s2, t2, h2, BB, H2, H1, H2, H2);
  gemm_wmma_f32_kernel<true ><<<(8 * (H3    / 16)) / 8, 256, 0, stream>>>(
      h2, W3,  s3, t3, h3, BB, H3, H2, H3, H3);
  gemm_wmma_f32_kernel<false><<<(8 * (NOUTP / 16)) / 8, 256, 0, stream>>>(
      h3, W4p, s4, t4, out, BB, NOUTP, H3, NOUT, NOUT);
}